// TopKLoss_67242007986762
// MI455X (gfx1250) — compile-verified
//
#include <hip/hip_runtime.h>
#include <hip/hip_bf16.h>
#include <math.h>

#define B_SZ     128
#define D_EMB    256
#define NSAMP    1000
#define KTOP     8
#define NSHAPES  8000

typedef __attribute__((ext_vector_type(2))) float v2f;
typedef __attribute__((ext_vector_type(8))) float v8f;

// ---------------------------------------------------------------------------
// JAX Threefry2x32 (exact: 20 rounds, 5 key-schedule injections)
// ---------------------------------------------------------------------------
__host__ __device__ __forceinline__ unsigned rotl32(unsigned x, int r) {
  return (x << r) | (x >> (32 - r));
}

__host__ __device__ __forceinline__ void threefry2x32(unsigned k0, unsigned k1,
                                                      unsigned x0, unsigned x1,
                                                      unsigned& y0, unsigned& y1) {
  const unsigned ks0 = k0, ks1 = k1, ks2 = k0 ^ k1 ^ 0x1BD11BDAu;
  x0 += ks0; x1 += ks1;
#define TF_R(r) { x0 += x1; x1 = rotl32(x1, (r)); x1 ^= x0; }
  TF_R(13) TF_R(15) TF_R(26) TF_R(6)
  x0 += ks1; x1 += ks2 + 1u;
  TF_R(17) TF_R(29) TF_R(16) TF_R(24)
  x0 += ks2; x1 += ks0 + 2u;
  TF_R(13) TF_R(15) TF_R(26) TF_R(6)
  x0 += ks0; x1 += ks1 + 3u;
  TF_R(17) TF_R(29) TF_R(16) TF_R(24)
  x0 += ks1; x1 += ks2 + 4u;
  TF_R(13) TF_R(15) TF_R(26) TF_R(6)
  x0 += ks2; x1 += ks0 + 5u;
#undef TF_R
  y0 = x0; y1 = x1;
}

// bits -> jax.random.normal sample (uniform in [nextafter(-1,0),1), sqrt(2)*erfinv)
__device__ __forceinline__ float normal_from_bits(unsigned bits) {
  float f = __uint_as_float((bits >> 9) | 0x3f800000u) - 1.0f;   // [0,1)
  const float mn = -0.99999994f;                                  // nextafter(-1,0)
  float u = fmaxf(mn, f * (1.0f - mn) + mn);
  float w = -log1pf(-u * u);
  float p;
  if (w < 5.0f) {
    w -= 2.5f;
    p = 2.81022636e-08f;
    p = fmaf(p, w, 3.43273939e-07f);
    p = fmaf(p, w, -3.5233877e-06f);
    p = fmaf(p, w, -4.39150654e-06f);
    p = fmaf(p, w, 0.00021858087f);
    p = fmaf(p, w, -0.00125372503f);
    p = fmaf(p, w, -0.00417768164f);
    p = fmaf(p, w, 0.246640727f);
    p = fmaf(p, w, 1.50140941f);
  } else {
    w = sqrtf(w) - 3.0f;
    p = -0.000200214257f;
    p = fmaf(p, w, 0.000100950558f);
    p = fmaf(p, w, 0.00134934322f);
    p = fmaf(p, w, -0.00367342844f);
    p = fmaf(p, w, 0.00573950773f);
    p = fmaf(p, w, -0.0076224613f);
    p = fmaf(p, w, 0.00943887047f);
    p = fmaf(p, w, 1.00167406f);
    p = fmaf(p, w, 2.83297682f);
  }
  return 1.41421356237f * p * u;
}

// ---------------------------------------------------------------------------
// Kernel 1: pred_sim = scan @ cad^T via V_WMMA_F32_16X16X4_F32.
// One wave per 16x16 output tile; 64 WMMA ops along K.
// A(16x4): lane holds M=lane&15, VGPR0=K {0|2}, VGPR1=K+1 (ISA 7.12.2).
// B(4x16): lane holds N=lane&15, same K striping (B[k][n] = cad[n][k]).
// ---------------------------------------------------------------------------
__global__ void __launch_bounds__(32)
predsim_wmma_kernel(const float* __restrict__ scan, const float* __restrict__ cad,
                    float* __restrict__ pred) {
  const int tile = blockIdx.x;              // 0..63
  const int m0 = (tile >> 3) << 4;
  const int n0 = (tile & 7) << 4;
  const int lane = threadIdx.x;             // 0..31, wave32
  const int row = lane & 15;
  const int khalf = (lane >> 4) << 1;       // 0 or 2
  const float* arow = scan + (m0 + row) * D_EMB + khalf;
  const float* brow = cad  + (n0 + row) * D_EMB + khalf;
  v8f c = {0.f, 0.f, 0.f, 0.f, 0.f, 0.f, 0.f, 0.f};
#pragma unroll 8
  for (int kb = 0; kb < D_EMB; kb += 4) {
    v2f a, b;
    a.x = arow[kb]; a.y = arow[kb + 1];
    b.x = brow[kb]; b.y = brow[kb + 1];
    c = __builtin_amdgcn_wmma_f32_16x16x4_f32(false, a, false, b, (short)0, c,
                                              false, false);
  }
  const int mbase = m0 + ((lane >> 4) << 3);  // C layout: lanes 16-31 -> M+8
#pragma unroll
  for (int r = 0; r < 8; ++r)
    pred[(mbase + r) * B_SZ + n0 + (lane & 15)] = c[r];
}

// ---------------------------------------------------------------------------
// Kernel 2: tgt_sim[i][j] = similarity_matrix[scannet_idx[i]][shapenet_idx[j]]
// ---------------------------------------------------------------------------
__global__ void __launch_bounds__(256)
tgtsim_gather_kernel(const float* __restrict__ simmat, const int* __restrict__ sidx,
                     const int* __restrict__ hidx, float* __restrict__ tgt) {
  int t = blockIdx.x * blockDim.x + threadIdx.x;
  if (t < B_SZ * B_SZ) {
    int i = t >> 7, j = t & 127;
    tgt[t] = simmat[(long)sidx[i] * NSHAPES + hidx[j]];
  }
}

// ---------------------------------------------------------------------------
// Kernel 3: perturbed top-k indicator counts.
// Noise flat index (i,s,d) pairs with (i+64,s,d) across Threefry halves, so a
// block owns row pair (rowLo, rowLo+64) and 250 samples; one thread = 1 sample,
// streaming d=0..127 with two register top-8 lists, then rank->LDS counters.
// ---------------------------------------------------------------------------
__global__ void __launch_bounds__(256)
topk_count_kernel(const float* __restrict__ sim, unsigned key0, unsigned key1,
                  float sigma, unsigned* __restrict__ gcnt) {
  __shared__ unsigned s_cnt[2][KTOP * B_SZ];   // 8 KB
  __shared__ float s_sim[2][B_SZ];
  const int rowLo = blockIdx.x >> 2;           // 0..63
  const int split = blockIdx.x & 3;

  for (int t = threadIdx.x; t < 2 * KTOP * B_SZ; t += blockDim.x)
    ((unsigned*)s_cnt)[t] = 0u;
  for (int t = threadIdx.x; t < 2 * B_SZ; t += blockDim.x)
    s_sim[t >> 7][t & 127] = sim[(rowLo + (t >> 7) * 64) * B_SZ + (t & 127)];
  __syncthreads();

  if (threadIdx.x < 250) {
    const int s = split * 250 + threadIdx.x;
    float v0[8], v1[8];
    int id0[8], id1[8];
#pragma unroll
    for (int j = 0; j < 8; ++j) { v0[j] = -INFINITY; v1[j] = -INFINITY; id0[j] = 0; id1[j] = 0; }
    float m0v = -INFINITY, m1v = -INFINITY;
    int m0i = 0, m1i = 0;
    const unsigned half = 64u * NSAMP * B_SZ;  // 8,192,000
    unsigned p = ((unsigned)rowLo * NSAMP + (unsigned)s) * B_SZ;
    for (int d = 0; d < B_SZ; ++d, ++p) {
      unsigned b0, b1;
      threefry2x32(key0, key1, p, p + half, b0, b1);
      float val0 = fmaf(sigma, normal_from_bits(b0), s_sim[0][d]);
      float val1 = fmaf(sigma, normal_from_bits(b1), s_sim[1][d]);
      if (val0 > m0v) {
        v0[m0i] = val0; id0[m0i] = d;
        m0v = v0[0]; m0i = 0;
#pragma unroll
        for (int j = 1; j < 8; ++j) if (v0[j] < m0v) { m0v = v0[j]; m0i = j; }
      }
      if (val1 > m1v) {
        v1[m1i] = val1; id1[m1i] = d;
        m1v = v1[0]; m1i = 0;
#pragma unroll
        for (int j = 1; j < 8; ++j) if (v1[j] < m1v) { m1v = v1[j]; m1i = j; }
      }
    }
    // rank by index (= position after jnp.sort of top-k indices), bump counters
#pragma unroll
    for (int a = 0; a < 8; ++a) {
      int r0 = 0, r1 = 0;
#pragma unroll
      for (int bq = 0; bq < 8; ++bq) {
        r0 += (id0[bq] < id0[a]);
        r1 += (id1[bq] < id1[a]);
      }
      atomicAdd(&s_cnt[0][r0 * B_SZ + id0[a]], 1u);
      atomicAdd(&s_cnt[1][r1 * B_SZ + id1[a]], 1u);
    }
  }
  __syncthreads();
  for (int t = threadIdx.x; t < 2 * KTOP * B_SZ; t += blockDim.x) {
    int h = t >> 10, idx = t & 1023;
    unsigned c = s_cnt[h][idx];
    if (c) atomicAdd(&gcnt[(rowLo + h * 64) * (KTOP * B_SZ) + idx], c);
  }
}

// ---------------------------------------------------------------------------
// Kernel 4: loss = -sum_{i,j,d} cntP*cntT*tgt_sim / (n^2 * k * b)
// ---------------------------------------------------------------------------
__global__ void __launch_bounds__(256)
loss_reduce_kernel(const unsigned* __restrict__ cntP, const unsigned* __restrict__ cntT,
                   const float* __restrict__ tgt, float* __restrict__ out) {
  __shared__ double s_acc[256];
  double acc = 0.0;
  for (int idx = threadIdx.x; idx < B_SZ * KTOP * B_SZ; idx += 256) {
    unsigned a = cntP[idx], b = cntT[idx];
    if (a && b) {
      int i = idx >> 10, d = idx & 127;
      acc += (double)a * (double)b * (double)tgt[i * B_SZ + d];
    }
  }
  s_acc[threadIdx.x] = acc;
  __syncthreads();
  for (int off = 128; off > 0; off >>= 1) {
    if (threadIdx.x < off) s_acc[threadIdx.x] += s_acc[threadIdx.x + off];
    __syncthreads();
  }
  if (threadIdx.x == 0) {
    const double scale = (double)NSAMP * (double)NSAMP * (double)KTOP * (double)B_SZ;
    out[0] = (float)(-s_acc[0] / scale);
  }
}

// ---------------------------------------------------------------------------
extern "C" void kernel_launch(void* const* d_in, const int* in_sizes, int n_in,
                              void* d_out, int out_size, void* d_ws, size_t ws_size,
                              hipStream_t stream) {
  const float* cad    = (const float*)d_in[0];   // (128, 256)
  const float* scan   = (const float*)d_in[1];   // (128, 256)
  const float* simmat = (const float*)d_in[2];   // (2000, 8000)
  const int*   sidx   = (const int*)d_in[3];     // (128,)
  const int*   hidx   = (const int*)d_in[4];     // (128,)
  float* out = (float*)d_out;

  // workspace layout
  float* predSim = (float*)d_ws;                         // 16384 f32
  float* tgtSim  = predSim + B_SZ * B_SZ;                // 16384 f32
  unsigned* cntP = (unsigned*)(tgtSim + B_SZ * B_SZ);    // 131072 u32
  unsigned* cntT = cntP + B_SZ * KTOP * B_SZ;            // 131072 u32

  // host-side: nk1, nk2 = jax.random.split(jax.random.key(42))
  // key(42) -> (0,42); split = threefry over counts [0,1,2,3] halved as (0,2),(1,3)
  unsigned a0, a1, b0, b1;
  threefry2x32(0u, 42u, 0u, 2u, a0, a1);
  threefry2x32(0u, 42u, 1u, 3u, b0, b1);
  const unsigned nk1_0 = a0, nk1_1 = b0;   // out[0], out[1]
  const unsigned nk2_0 = a1, nk2_1 = b1;   // out[2], out[3]

  hipMemsetAsync(cntP, 0, sizeof(unsigned) * 2u * B_SZ * KTOP * B_SZ, stream);

  predsim_wmma_kernel<<<64, 32, 0, stream>>>(scan, cad, predSim);
  tgtsim_gather_kernel<<<64, 256, 0, stream>>>(simmat, sidx, hidx, tgtSim);
  topk_count_kernel<<<256, 256, 0, stream>>>(predSim, nk1_0, nk1_1, 0.05f, cntP);
  topk_count_kernel<<<256, 256, 0, stream>>>(tgtSim, nk2_0, nk2_1, 0.05f / 200.0f, cntT);
  loss_reduce_kernel<<<1, 256, 0, stream>>>(cntP, cntT, tgtSim, out);
}